// LLMCRec_81982335746485
// MI455X (gfx1250) — compile-verified
//
#include <hip/hip_runtime.h>
#include <math.h>

#define DFEAT 64
#define DF2   32   // float2 elements per feature row
#define WPB   8    // waves per block (256 threads / wave32)
#define CHUNK 8    // neighbor rows staged per async pipeline stage

// Raw LDS byte offset of a pointer to __shared__ memory (for async-to-LDS VDST).
__device__ __forceinline__ unsigned lds_off(const void* p) {
  return (unsigned)(uintptr_t)(__attribute__((address_space(3))) const char*)(p);
}

// ------------------------- utility kernels -------------------------

__global__ void k_zero_i32(int* __restrict__ p, int n) {
  int i = blockIdx.x * blockDim.x + threadIdx.x;
  if (i < n) p[i] = 0;
}

__global__ void k_zero_f32(float* __restrict__ p, long long n) {
  long long i = (long long)blockIdx.x * blockDim.x + threadIdx.x;
  if (i < n) p[i] = 0.f;
}

__global__ void k_copy_f4(const float4* __restrict__ src, float4* __restrict__ dst,
                          long long n4) {
  long long i = (long long)blockIdx.x * blockDim.x + threadIdx.x;
  if (i < n4) dst[i] = src[i];
}

// dst[0:a4+b4] = concat(A,B) as float4; accumulate==1 -> dst += concat(A,B)
__global__ void k_concat(const float4* __restrict__ A, long long a4,
                         const float4* __restrict__ B, long long b4,
                         float4* __restrict__ dst, int accumulate) {
  long long i = (long long)blockIdx.x * blockDim.x + threadIdx.x;
  if (i >= a4 + b4) return;
  float4 v = (i < a4) ? A[i] : B[i - a4];
  if (accumulate) {
    float4 d = dst[i];
    d.x += v.x; d.y += v.y; d.z += v.z; d.w += v.w;
    dst[i] = d;
  } else {
    dst[i] = v;
  }
}

// ------------------------- CSR construction -------------------------

__global__ void k_count(const int* __restrict__ rows, int* __restrict__ counts, int E) {
  int i = blockIdx.x * blockDim.x + threadIdx.x;
  if (i < E) atomicAdd(&counts[rows[i]], 1);
}

// single-workgroup exclusive scan: counts[0..N-1] -> row_ptr[0..N], cursor[0..N-1]
__global__ __launch_bounds__(1024) void k_scan(const int* __restrict__ counts,
                                               int* __restrict__ row_ptr,
                                               int* __restrict__ cursor, int N) {
  __shared__ int tsum[1024];
  const int tid = threadIdx.x;
  const int chunk = (N + 1023) / 1024;
  const int begin = tid * chunk;
  const int endi  = (begin + chunk < N) ? (begin + chunk) : N;
  int s = 0;
  for (int i = begin; i < endi; ++i) s += counts[i];
  tsum[tid] = s;
  __syncthreads();
  // Hillis-Steele inclusive scan over 1024 partials
  for (int off = 1; off < 1024; off <<= 1) {
    int v = (tid >= off) ? tsum[tid - off] : 0;
    __syncthreads();
    tsum[tid] += v;
    __syncthreads();
  }
  int run = (tid == 0) ? 0 : tsum[tid - 1];  // exclusive prefix of this chunk
  for (int i = begin; i < endi; ++i) {
    row_ptr[i] = run;
    cursor[i]  = run;
    run += counts[i];
  }
  if (tid == 1023) row_ptr[N] = tsum[1023];
}

__global__ void k_scatter(const int* __restrict__ rows, const int* __restrict__ cols,
                          const float* __restrict__ vals, int* __restrict__ cursor,
                          int* __restrict__ csr_col, float* __restrict__ csr_val, int E) {
  int i = blockIdx.x * blockDim.x + threadIdx.x;
  if (i < E) {
    int r = rows[i];
    int p = atomicAdd(&cursor[r], 1);
    csr_col[p] = cols[i];
    csr_val[p] = vals[i];
  }
}

// ------------------------- fused SpMM layer -------------------------
// One wave32 per output row, neighbor feature rows staged into LDS with a
// double-buffered GLOBAL_LOAD_ASYNC_TO_LDS_B128 pipeline (ASYNCcnt-tracked).
// newfeat[row] = scale * sum_j val_j * feat[col_j];
// acc[row]   += newfeat[row] / max(||newfeat[row]||, 1e-12).
__global__ __launch_bounds__(256) void k_spmm(const int* __restrict__ row_ptr,
                                              const int* __restrict__ csr_col,
                                              const float* __restrict__ csr_val,
                                              const float* __restrict__ feat,
                                              float* __restrict__ newfeat,
                                              float* __restrict__ acc,
                                              float scale, int N) {
  // per-wave double buffer: 2 x (CHUNK rows x 256B) = 4KB/wave, 32KB/WG
  __shared__ __attribute__((aligned(16))) float s_stage[WPB][2][CHUNK * DFEAT];

  const int wslot = threadIdx.x >> 5;
  const int wid   = blockIdx.x * WPB + wslot;
  const int lane  = threadIdx.x & 31;
  if (wid >= N) return;

  const int start = row_ptr[wid];
  const int end   = row_ptr[wid + 1];
  const int deg   = end - start;

  float ax = 0.f, ay = 0.f;

  if (deg > 0) {
    const int nchunks = (deg + CHUNK - 1) / CHUNK;
    const unsigned long long fbase = (unsigned long long)(uintptr_t)feat;
    const int      sub  = lane >> 4;                    // which of 2 rows per async op
    const unsigned lofs = (unsigned)(lane & 15) * 16u;  // byte offset within 256B row

    // stage chunk t (8 neighbor rows) into buffer b with 4 async b128 ops
    auto issue = [&](int t, int b) {
      int idx = start + t * CHUNK + lane;
      int clamped = (idx < end) ? idx : (end - 1);  // keep addresses legal
      int c = 0;
      if (lane < CHUNK) c = __ldg(&csr_col[clamped]);
      const unsigned lbase = lds_off(&s_stage[wslot][b][0]);
#pragma unroll
      for (int j = 0; j < CHUNK / 2; ++j) {
        int cj = __shfl(c, 2 * j + sub, 32);
        unsigned long long g = fbase + ((unsigned long long)(unsigned)cj << 8) + lofs;
        unsigned l = lbase + (unsigned)(2 * j + sub) * 256u + lofs;
        asm volatile("global_load_async_to_lds_b128 %0, %1, off"
                     :: "v"(l), "v"(g)
                     : "memory");
      }
    };

    issue(0, 0);
    for (int t = 0; t < nchunks; ++t) {
      const int b = t & 1;
      if (t + 1 < nchunks) {
        issue(t + 1, b ^ 1);
        // 8 async ops outstanding; in-order completion => <=4 means chunk t landed
        asm volatile("s_wait_asynccnt 0x4" ::: "memory");
      } else {
        asm volatile("s_wait_asynccnt 0x0" ::: "memory");
      }
      // consume staged chunk t from LDS
      int idx = start + t * CHUNK + lane;
      float w = 0.f;
      if (lane < CHUNK && idx < end) w = __ldg(&csr_val[idx]);
      int cnt = deg - t * CHUNK; if (cnt > CHUNK) cnt = CHUNK;
      const float2* srow = (const float2*)&s_stage[wslot][b][0];
      for (int g = 0; g < cnt; ++g) {
        float ww = __shfl(w, g, 32);
        float2 f = srow[g * DF2 + lane];
        ax = fmaf(ww, f.x, ax);
        ay = fmaf(ww, f.y, ay);
      }
    }
  }

  ax *= scale; ay *= scale;

  const long long o = (long long)wid * DF2 + lane;
  ((float2*)newfeat)[o] = make_float2(ax, ay);

  // wave-level sum of squares for the row L2 norm
  float ss = ax * ax + ay * ay;
#pragma unroll
  for (int off = 16; off > 0; off >>= 1) ss += __shfl_xor(ss, off, 32);
  const float inv = 1.f / fmaxf(sqrtf(ss), 1e-12f);

  float2* acc2 = (float2*)acc;
  float2 a = acc2[o];
  a.x = fmaf(ax, inv, a.x);
  a.y = fmaf(ay, inv, a.y);
  acc2[o] = a;
}

// ------------------------- concept->course aggregation -------------------------
// out[r*64 + :] += v * kfeat[c*64 + :], edge-parallel, wave32 per edge (2 floats/lane).
__global__ void k_ca_spmm(const int* __restrict__ rows, const int* __restrict__ cols,
                          const float* __restrict__ vals, const float* __restrict__ kfeat,
                          float* __restrict__ out, int E) {
  long long t = (long long)blockIdx.x * blockDim.x + threadIdx.x;
  long long e = t >> 5;
  int lane = (int)(t & 31);
  if (e >= E) return;
  int r = rows[e], c = cols[e];
  float v = vals[e];
  float2 f = ((const float2*)kfeat)[(long long)c * DF2 + lane];
  atomicAdd(&out[(long long)r * DFEAT + lane * 2    ], v * f.x);
  atomicAdd(&out[(long long)r * DFEAT + lane * 2 + 1], v * f.y);
}

// ------------------------- host driver -------------------------

extern "C" void kernel_launch(void* const* d_in, const int* in_sizes, int n_in,
                              void* d_out, int out_size, void* d_ws, size_t ws_size,
                              hipStream_t stream) {
  (void)n_in; (void)out_size; (void)ws_size;

  const float* learners = (const float*)d_in[0];
  const float* courses  = (const float*)d_in[1];
  const float* concepts = (const float*)d_in[2];
  const int*   cg_rows  = (const int*)d_in[3];
  const int*   cg_cols  = (const int*)d_in[4];
  const float* cg_vals  = (const float*)d_in[5];
  const int*   kg_rows  = (const int*)d_in[6];
  const int*   kg_cols  = (const int*)d_in[7];
  const float* kg_vals  = (const float*)d_in[8];
  const int*   ca_rows  = (const int*)d_in[9];
  const int*   ca_cols  = (const int*)d_in[10];
  const float* ca_vals  = (const float*)d_in[11];

  const int L = in_sizes[0] / DFEAT;
  const int C = in_sizes[1] / DFEAT;
  const int K = in_sizes[2] / DFEAT;
  const int E_cg = in_sizes[3];
  const int E_kg = in_sizes[6];
  const int E_ca = in_sizes[9];
  const int N_cg = L + C;
  const int N_kg = L + K;
  const int maxN = (N_cg > N_kg) ? N_cg : N_kg;
  const int maxE = (E_cg > E_kg) ? E_cg : E_kg;

  // ---- workspace carve (aligned 256B) ----
  char* wsp = (char*)d_ws;
  auto carve = [&](size_t bytes) -> char* {
    char* p = wsp;
    wsp += (bytes + 255) & ~(size_t)255;
    return p;
  };
  float* featA   = (float*)carve((size_t)maxN * DFEAT * sizeof(float));
  float* featB   = (float*)carve((size_t)maxN * DFEAT * sizeof(float));
  float* accK    = (float*)carve((size_t)N_kg * DFEAT * sizeof(float));
  int*   row_ptr = (int*)  carve((size_t)(maxN + 1) * sizeof(int));
  int*   counts  = (int*)  carve((size_t)maxN * sizeof(int));
  int*   cursor  = (int*)  carve((size_t)maxN * sizeof(int));
  int*   csr_col = (int*)  carve((size_t)maxE * sizeof(int));
  float* csr_val = (float*)carve((size_t)maxE * sizeof(float));

  float* out = (float*)d_out;

  auto csr_build = [&](const int* rows, const int* cols, const float* vals,
                       int E, int N) {
    k_zero_i32<<<(N + 255) / 256, 256, 0, stream>>>(counts, N);
    k_count<<<(E + 255) / 256, 256, 0, stream>>>(rows, counts, E);
    k_scan<<<1, 1024, 0, stream>>>(counts, row_ptr, cursor, N);
    k_scatter<<<(E + 255) / 256, 256, 0, stream>>>(rows, cols, vals, cursor,
                                                   csr_col, csr_val, E);
  };

  // Two-layer LightGCN propagation; acc may alias d_out (accumulateInit=1).
  auto propagate = [&](const float* Af, int nA, const float* Bf, int nB,
                       const int* rows, const int* cols, const float* vals, int E,
                       float* acc, int accumulateInit) {
    const int N = nA + nB;
    csr_build(rows, cols, vals, E, N);
    const long long a4 = (long long)nA * (DFEAT / 4);
    const long long b4 = (long long)nB * (DFEAT / 4);
    const long long tot4 = a4 + b4;
    const int g = (int)((tot4 + 255) / 256);
    k_concat<<<g, 256, 0, stream>>>((const float4*)Af, a4, (const float4*)Bf, b4,
                                    (float4*)featA, 0);
    k_concat<<<g, 256, 0, stream>>>((const float4*)Af, a4, (const float4*)Bf, b4,
                                    (float4*)acc, accumulateInit);
    const int gs = (N + WPB - 1) / WPB;
    k_spmm<<<gs, 256, 0, stream>>>(row_ptr, csr_col, csr_val, featA, featB, acc,
                                   0.5f, N);
    k_spmm<<<gs, 256, 0, stream>>>(row_ptr, csr_col, csr_val, featB, featA, acc,
                                   1.f / 3.f, N);
  };

  // 1) concept-grained view -> accK  (rows [0:L]=kg_l, [L:L+K]=kg_k)
  propagate(learners, L, concepts, K, kg_rows, kg_cols, kg_vals, E_kg, accK, 0);

  // 2) d_out[0:L] = kg_l ; d_out[L:L+C] = 0
  {
    const long long n4 = (long long)L * (DFEAT / 4);
    k_copy_f4<<<(int)((n4 + 255) / 256), 256, 0, stream>>>((const float4*)accK,
                                                           (float4*)out, n4);
    const long long nz = (long long)C * DFEAT;
    k_zero_f32<<<(int)((nz + 255) / 256), 256, 0, stream>>>(
        out + (long long)L * DFEAT, nz);
  }

  // 3) kg_c: aggregate concept rows of accK into course rows of d_out
  {
    const long long t = (long long)E_ca * 32;
    k_ca_spmm<<<(int)((t + 255) / 256), 256, 0, stream>>>(
        ca_rows, ca_cols, ca_vals,
        accK + (long long)L * DFEAT, out + (long long)L * DFEAT, E_ca);
  }

  // 4) course-grained view, accumulated directly into d_out
  propagate(learners, L, courses, C, cg_rows, cg_cols, cg_vals, E_cg, out, 1);
}